// ReactionMPNN_62732292326000
// MI455X (gfx1250) — compile-verified
//
#include <hip/hip_runtime.h>
#include <hip/hip_bf16.h>

typedef __attribute__((ext_vector_type(2))) float v2f;
typedef __attribute__((ext_vector_type(8))) float v8f;

#define HDIM 128
#define FCDIM 64

// ---------------------------------------------------------------------------
// zero-fill (graph-capture safe, no hipMemsetAsync dependence)
// ---------------------------------------------------------------------------
__global__ __launch_bounds__(256)
void zero_kernel(float* __restrict__ p, int n) {
    int i = blockIdx.x * 256 + threadIdx.x;
    if (i < n) p[i] = 0.0f;
}

// ---------------------------------------------------------------------------
// Fused NNConv message kernel.
//   For a tile of 16 edges: per (i, o-block) compute the 16x16 slice of
//   relu(ea @ We + be) with V_WMMA_F32_16X16X4_F32 (K = F_EDGE = 4),
//   multiply by gathered x_src[:, i] (staged in LDS) and accumulate msg,
//   then scatter-add to agg[dst] with f32 atomics. Per-edge weight matrices
//   never touch HBM.
// One wave (32 threads) per block; EXEC all ones as WMMA requires.
// ---------------------------------------------------------------------------
template<int DIN>
__global__ __launch_bounds__(32)
void edge_msg_kernel(const float* __restrict__ xin,   // [N, DIN]
                     const int*   __restrict__ src,   // [E]
                     const int*   __restrict__ dst,   // [E]
                     const float* __restrict__ ea,    // [E, 4]
                     const float* __restrict__ We,    // [4, DIN*HDIM]
                     const float* __restrict__ be,    // [DIN*HDIM]
                     float*       __restrict__ agg,   // [N, HDIM] accumulated
                     int E_)
{
    __shared__ float xs[DIN * 16];          // xs[i*16 + e]
    const int lane = threadIdx.x;           // 0..31 (wave32)
    const int lm   = lane & 15;
    const int half = lane >> 4;             // 0 / 1
    const int e0   = blockIdx.x * 16;

    // destinations for the 8 C/D rows this lane owns (row m = v + 8*half)
    int dsts[8];
#pragma unroll
    for (int v = 0; v < 8; ++v) {
        int e = e0 + v + 8 * half;
        dsts[v] = (e < E_) ? dst[e] : -1;
    }

    // A operand: 16x4 edge-attr tile.
    // Layout: lane m = lane&15; VGPR0 holds K = 2*half, VGPR1 holds K = 2*half+1.
    v2f A;
    {
        int e = e0 + lm;
        if (e < E_) { A.x = ea[e * 4 + 2 * half]; A.y = ea[e * 4 + 2 * half + 1]; }
        else        { A.x = 0.0f;                 A.y = 0.0f; }
    }

    // Stage gathered x_src tile into LDS, [i][e] layout (zero rows for pad
    // edges -> their messages are exactly zero regardless of weights).
    for (int e = 0; e < 16; ++e) {
        int ge = e0 + e;
        int s  = (ge < E_) ? src[ge] : -1;         // uniform -> scalar load
        for (int i = lane; i < DIN; i += 32)
            xs[i * 16 + e] = (s >= 0) ? xin[s * DIN + i] : 0.0f;
    }
    __syncthreads();

    const int JD = DIN * HDIM;                     // row stride of We

    for (int ob = 0; ob < 8; ++ob) {               // 8 blocks of 16 outputs
        v8f acc;
#pragma unroll
        for (int q = 0; q < 8; ++q) acc[q] = 0.0f;

#pragma unroll 4
        for (int i = 0; i < DIN; ++i) {
            int j = i * HDIM + ob * 16 + lm;       // flattened (i,o) column
            // B operand 4x16: VGPR0 rows {0,2}, VGPR1 rows {1,3} by lane half
            v2f B;
            B.x = We[(2 * half)     * JD + j];
            B.y = We[(2 * half + 1) * JD + j];
            // bias broadcast along M via the C operand
            float bv = be[j];
            v8f c;
#pragma unroll
            for (int q = 0; q < 8; ++q) c[q] = bv;

            v8f d = __builtin_amdgcn_wmma_f32_16x16x4_f32(
                false, A, false, B, (short)0, c, false, false);

            // relu, multiply by x_src[row, i], accumulate message
            const float* xr = &xs[i * 16 + 8 * half];  // 8 consecutive floats
#pragma unroll
            for (int v = 0; v < 8; ++v) {
                float w = fmaxf(d[v], 0.0f);
                acc[v] = fmaf(xr[v], w, acc[v]);
            }
        }

        // scatter-add into agg[dst]
        int o = ob * 16 + lm;
#pragma unroll
        for (int v = 0; v < 8; ++v)
            if (dsts[v] >= 0)
                atomicAdd(&agg[dsts[v] * HDIM + o], acc[v]);
    }
}

// ---------------------------------------------------------------------------
// Node update: h = relu(BN(agg + x @ Wr + br)).  x@Wr via the same f32 WMMA,
// K-loop in steps of 4 with zero padding (handles DIN=22).
// Block = 1 wave; grid = nodeTiles * 8 o-blocks.
// ---------------------------------------------------------------------------
template<int DIN>
__global__ __launch_bounds__(32)
void node_update_kernel(const float* __restrict__ agg,  // [N, HDIM]
                        const float* __restrict__ xin,  // [N, DIN]
                        const float* __restrict__ Wr,   // [DIN, HDIM]
                        const float* __restrict__ br,
                        const float* __restrict__ gg,
                        const float* __restrict__ bb,
                        const float* __restrict__ rm,
                        const float* __restrict__ rv,
                        float*       __restrict__ hout, // [N, HDIM]
                        int Nn)
{
    const int lane = threadIdx.x;
    const int lm   = lane & 15;
    const int half = lane >> 4;
    const int tile = blockIdx.x >> 3;
    const int ob   = blockIdx.x & 7;
    const int n0   = tile * 16;
    const int o    = ob * 16 + lm;

    v8f d;
#pragma unroll
    for (int q = 0; q < 8; ++q) d[q] = 0.0f;

    const int node_a = n0 + lm;
    const bool nok   = node_a < Nn;

    for (int k0 = 0; k0 < DIN; k0 += 4) {
        int ka = k0 + 2 * half;
        v2f A;
        A.x = (nok && ka     < DIN) ? xin[node_a * DIN + ka]     : 0.0f;
        A.y = (nok && ka + 1 < DIN) ? xin[node_a * DIN + ka + 1] : 0.0f;
        v2f B;
        B.x = (ka     < DIN) ? Wr[(ka)     * HDIM + o] : 0.0f;
        B.y = (ka + 1 < DIN) ? Wr[(ka + 1) * HDIM + o] : 0.0f;
        d = __builtin_amdgcn_wmma_f32_16x16x4_f32(
            false, A, false, B, (short)0, d, false, false);
    }

    const float gv  = gg[o];
    const float bv  = bb[o];
    const float mv  = rm[o];
    const float inv = rsqrtf(rv[o] + 1e-5f);
    const float brv = br[o];

#pragma unroll
    for (int v = 0; v < 8; ++v) {
        int node = n0 + v + 8 * half;
        if (node < Nn) {
            float val = d[v] + agg[node * HDIM + o] + brv;
            val = (val - mv) * inv * gv + bv;
            hout[node * HDIM + o] = fmaxf(val, 0.0f);
        }
    }
}

// ---------------------------------------------------------------------------
// Global mean pool (accumulate phase): one block per node.
// ---------------------------------------------------------------------------
__global__ __launch_bounds__(HDIM)
void pool_kernel(const float* __restrict__ h, const int* __restrict__ batch,
                 float* __restrict__ sums, float* __restrict__ cnt, int Nn)
{
    int n = blockIdx.x;
    if (n >= Nn) return;
    int t = threadIdx.x;
    int g = batch[n];
    atomicAdd(&sums[g * HDIM + t], h[n * HDIM + t]);
    if (t == 0) atomicAdd(&cnt[g], 1.0f);
}

// ---------------------------------------------------------------------------
// Head MLP: out[g] = relu(pooled @ Wf1 + bf1) @ Wf2 + bf2.  Tiny (100x128x64).
// ---------------------------------------------------------------------------
__global__ __launch_bounds__(FCDIM)
void head_kernel(const float* __restrict__ sums, const float* __restrict__ cnt,
                 const float* __restrict__ Wf1, const float* __restrict__ bf1,
                 const float* __restrict__ Wf2, const float* __restrict__ bf2,
                 float* __restrict__ out)
{
    __shared__ float red[FCDIM];
    int g = blockIdx.x;
    int t = threadIdx.x;
    float invc = 1.0f / fmaxf(cnt[g], 1.0f);
    float s = 0.0f;
    for (int o = 0; o < HDIM; ++o)
        s = fmaf(sums[g * HDIM + o], Wf1[o * FCDIM + t], s);
    s = fmaxf(fmaf(s, invc, 0.0f) + bf1[t], 0.0f) * Wf2[t];
    red[t] = s;
    __syncthreads();
    if (t == 0) {
        float a = bf2[0];
        for (int i = 0; i < FCDIM; ++i) a += red[i];
        out[g] = a;
    }
}

// ---------------------------------------------------------------------------
extern "C" void kernel_launch(void* const* d_in, const int* in_sizes, int n_in,
                              void* d_out, int out_size, void* d_ws, size_t ws_size,
                              hipStream_t stream)
{
    const float* x    = (const float*)d_in[0];
    const int*   ei   = (const int*)  d_in[1];   // [2, E]
    const float* ea   = (const float*)d_in[2];   // [E, 4]
    const int*   bat  = (const int*)  d_in[3];   // [N]
    const float* We1  = (const float*)d_in[4];  const float* be1 = (const float*)d_in[5];
    const float* We2  = (const float*)d_in[6];  const float* be2 = (const float*)d_in[7];
    const float* We3  = (const float*)d_in[8];  const float* be3 = (const float*)d_in[9];
    const float* Wr1  = (const float*)d_in[10]; const float* br1 = (const float*)d_in[11];
    const float* Wr2  = (const float*)d_in[12]; const float* br2 = (const float*)d_in[13];
    const float* Wr3  = (const float*)d_in[14]; const float* br3 = (const float*)d_in[15];
    const float* g1 = (const float*)d_in[16]; const float* b1 = (const float*)d_in[17];
    const float* rm1= (const float*)d_in[18]; const float* rv1= (const float*)d_in[19];
    const float* g2 = (const float*)d_in[20]; const float* b2 = (const float*)d_in[21];
    const float* rm2= (const float*)d_in[22]; const float* rv2= (const float*)d_in[23];
    const float* g3 = (const float*)d_in[24]; const float* b3 = (const float*)d_in[25];
    const float* rm3= (const float*)d_in[26]; const float* rv3= (const float*)d_in[27];
    const float* Wf1= (const float*)d_in[28]; const float* bf1= (const float*)d_in[29];
    const float* Wf2= (const float*)d_in[30]; const float* bf2= (const float*)d_in[31];

    const int N = in_sizes[3];
    const int E = in_sizes[2] / 4;
    const int G = out_size;
    const int FN = in_sizes[0] / N;   // 22 (template param must match)
    (void)FN; (void)n_in; (void)ws_size;

    const int* src = ei;
    const int* dstp = ei + E;

    // workspace partition (floats)
    float* ws   = (float*)d_ws;
    float* agg  = ws;                         // N*HDIM
    float* h1   = ws + (size_t)N * HDIM;      // N*HDIM
    float* h2   = ws + (size_t)2 * N * HDIM;  // N*HDIM
    float* sums = ws + (size_t)3 * N * HDIM;  // G*HDIM
    float* cnt  = sums + (size_t)G * HDIM;    // G

    const int NH = N * HDIM;
    const int eTiles = (E + 15) / 16;
    const int nTiles = (N + 15) / 16;
    dim3 wave(32);

    // ---- layer 1: x[N,22] -> h1[N,128] ----
    zero_kernel<<<(NH + 255) / 256, 256, 0, stream>>>(agg, NH);
    edge_msg_kernel<22><<<eTiles, wave, 0, stream>>>(x, src, dstp, ea, We1, be1, agg, E);
    node_update_kernel<22><<<nTiles * 8, wave, 0, stream>>>(agg, x, Wr1, br1,
                                                            g1, b1, rm1, rv1, h1, N);
    // ---- layer 2: h1 -> h2 ----
    zero_kernel<<<(NH + 255) / 256, 256, 0, stream>>>(agg, NH);
    edge_msg_kernel<128><<<eTiles, wave, 0, stream>>>(h1, src, dstp, ea, We2, be2, agg, E);
    node_update_kernel<128><<<nTiles * 8, wave, 0, stream>>>(agg, h1, Wr2, br2,
                                                             g2, b2, rm2, rv2, h2, N);
    // ---- layer 3: h2 -> h1 (reuse buffer) ----
    zero_kernel<<<(NH + 255) / 256, 256, 0, stream>>>(agg, NH);
    edge_msg_kernel<128><<<eTiles, wave, 0, stream>>>(h2, src, dstp, ea, We3, be3, agg, E);
    node_update_kernel<128><<<nTiles * 8, wave, 0, stream>>>(agg, h2, Wr3, br3,
                                                             g3, b3, rm3, rv3, h1, N);
    // ---- pool + head ----
    zero_kernel<<<(G * HDIM + G + 255) / 256, 256, 0, stream>>>(sums, G * HDIM + G);
    pool_kernel<<<N, HDIM, 0, stream>>>(h1, bat, sums, cnt, N);
    head_kernel<<<G, FCDIM, 0, stream>>>(sums, cnt, Wf1, bf1, Wf2, bf2, (float*)d_out);
}